// Rule_10471130267814
// MI455X (gfx1250) — compile-verified
//
#include <hip/hip_runtime.h>
#include <hip/hip_bf16.h>
#include <stdint.h>

// ---- problem constants (must match reference) ----
#define RADIUS   2
#define KK       5
#define K2       25
#define KCENTER  12
#define HH       768
#define WW       768
#define NN_      (HH * WW)          // 589824
#define DECAY    0.2f
#define INTEG    4.0f
#define REFRACT  5.0f
#define MIN_E    0.1f
#define TARGET_RATE 10.0f
#define PLASTIC_LR  0.0001f

// ---- tiling ----
#define TILE_W   32                  // one wave32 per row
#define TILE_H   8                   // 256 threads = 8 waves
#define HALO_W   (TILE_W + 2*RADIUS) // 36
#define HALO_H   (TILE_H + 2*RADIUS) // 12
#define HALO_N   (HALO_W * HALO_H)   // 432

#if defined(__has_builtin)
#if __has_builtin(__builtin_amdgcn_global_load_async_to_lds_b32)
#define USE_ASYNC_LDS 1
#endif
#endif

// streaming (read-once / write-once) accessors -> non-temporal cache policy
__device__ __forceinline__ float ld_nt(const float* p) {
    return __builtin_nontemporal_load(p);
}
__device__ __forceinline__ void st_nt(float* p, float v) {
    __builtin_nontemporal_store(v, p);
}

__global__ __launch_bounds__(256) void snn_step_kernel(
    const float* __restrict__ x,      // 6*N : S,V,R,A,T,E
    const float* __restrict__ noise,  // N
    const float* __restrict__ nn,     // 25*N  (k-major, pixel-contiguous)
    const float* __restrict__ EI,     // N
    const float* __restrict__ kI,     // N  (k_sums_I)
    const float* __restrict__ kE,     // N  (k_sums_E)
    float* __restrict__ out)          // 7*N (z) then 25*N (nn_new)
{
    __shared__ float ldsS [HALO_N];
    __shared__ float ldsA [HALO_N];
    __shared__ float ldsEI[HALO_N];

    const int tid = threadIdx.x;
    const int tx  = tid & (TILE_W - 1);
    const int ty  = tid >> 5;
    const int bx0 = blockIdx.x * TILE_W;
    const int by0 = blockIdx.y * TILE_H;

    const float* Sf = x;
    const float* Vf = x + 1 * NN_;
    const float* Rf = x + 2 * NN_;
    const float* Af = x + 3 * NN_;
    const float* Tf = x + 4 * NN_;
    const float* Ef = x + 5 * NN_;

    // ---- stage halo tiles of oldS / oldA / EI into LDS (wrap addressing) ----
    // These three fields have 25x reuse -> default (RT) policy + async DMA to LDS.
    for (int i = tid; i < HALO_N; i += 256) {
        int hy = i / HALO_W;
        int hx = i - hy * HALO_W;
        int gy = by0 + hy - RADIUS; if (gy < 0) gy += HH; else if (gy >= HH) gy -= HH;
        int gx = bx0 + hx - RADIUS; if (gx < 0) gx += WW; else if (gx >= WW) gx -= WW;
        int s  = gy * WW + gx;
#if defined(USE_ASYNC_LDS)
        __builtin_amdgcn_global_load_async_to_lds_b32((int*)(Sf + s), (int*)&ldsS [i], 0, 0);
        __builtin_amdgcn_global_load_async_to_lds_b32((int*)(Af + s), (int*)&ldsA [i], 0, 0);
        __builtin_amdgcn_global_load_async_to_lds_b32((int*)(EI + s), (int*)&ldsEI[i], 0, 0);
#else
        ldsS [i] = Sf[s];
        ldsA [i] = Af[s];
        ldsEI[i] = EI[s];
#endif
    }
#if defined(USE_ASYNC_LDS)
#if __has_builtin(__builtin_amdgcn_s_wait_asynccnt)
    __builtin_amdgcn_s_wait_asynccnt(0);
#else
    asm volatile("s_wait_asynccnt 0" ::: "memory");
#endif
#endif
    __syncthreads();

    const int gx = bx0 + tx;
    const int gy = by0 + ty;
    const int p  = gy * WW + gx;
    const int hc = (ty + RADIUS) * HALO_W + (tx + RADIUS);

    // ---- phase 1: synaptic input I = sum_k (S*EI)(nbr k) * nn[k,p] ----
    float nnv[K2];
    float I = 0.0f;
    #pragma unroll
    for (int dy = 0; dy < KK; ++dy) {
        #pragma unroll
        for (int dx = 0; dx < KK; ++dx) {
            const int k = dy * KK + dx;
            nnv[k] = ld_nt(nn + (size_t)k * NN_ + p);   // read-once 59 MB stream
            const int h = (ty + dy) * HALO_W + (tx + dx);
            I += ldsS[h] * ldsEI[h] * nnv[k];
        }
    }

    // ---- neuron state update (all read-once per-pixel streams) ----
    const float V0  = ld_nt(Vf + p);
    const float R0  = ld_nt(Rf + p);
    const float A0  = ldsA[hc];
    const float Tv  = ld_nt(Tf + p);
    const float Ev  = ld_nt(Ef + p);
    const float eic = ldsEI[hc];

    float V = V0 - DECAY * V0 + INTEG * I;   // CONST_I == 0
    const float S  = ((V > Tv) && (R0 > REFRACT) && (Ev > MIN_E)) ? 1.0f : 0.0f;
    const float Rn = (R0 + 1.0f) * (1.0f - S);
    const float An = A0 - A0 * 0.01f + S;
    const float Vn = V * (1.0f - S);

    // ---- phase 2a: plasticity deltas -> per-pixel I/E sums ----
    float sumI = 0.0f, sumE = 0.0f;
    #pragma unroll
    for (int dy = 0; dy < KK; ++dy) {
        #pragma unroll
        for (int dx = 0; dx < KK; ++dx) {
            const int k = dy * KK + dx;
            const int h = (ty + dy) * HALO_W + (tx + dx);
            const float preS = ldsS[h];
            const float preA = ldsA[h];
            const bool  inh  = (ldsEI[h] < 0.0f);
            const float common = S * preA;                      // post_spike * pre
            const float delta  = inh ? (preS * (An - TARGET_RATE) + common)
                                     : (common - preS * An);
            float nk = fmaxf(nnv[k] + PLASTIC_LR * delta, 0.0f);
            if (k == KCENTER) nk = 0.0f;
            if (inh) sumI += nk; else sumE += nk;
        }
    }
    const float scI = ld_nt(kI + p) / (sumI + 1e-6f);
    const float scE = 0.5f * ld_nt(kE + p) / (sumE + 1e-6f);

    // ---- phase 2b: recompute nk, normalize, EMA, stream out nn_new ----
    float* __restrict__ onn = out + 7 * NN_;
    #pragma unroll
    for (int dy = 0; dy < KK; ++dy) {
        #pragma unroll
        for (int dx = 0; dx < KK; ++dx) {
            const int k = dy * KK + dx;
            const int h = (ty + dy) * HALO_W + (tx + dx);
            const float preS = ldsS[h];
            const float preA = ldsA[h];
            const bool  inh  = (ldsEI[h] < 0.0f);
            const float common = S * preA;
            const float delta  = inh ? (preS * (An - TARGET_RATE) + common)
                                     : (common - preS * An);
            float nk = fmaxf(nnv[k] + PLASTIC_LR * delta, 0.0f);
            if (k == KCENTER) nk = 0.0f;
            const float comb = inh ? (nk * scI) : (nk * scE);
            st_nt(onn + (size_t)k * NN_ + p, 0.9f * nnv[k] + 0.1f * comb);  // write-once 59 MB stream
        }
    }

    // ---- z outputs: [S+noise, Vn, Rn, An, T, E, EI] ----
    st_nt(out + 0 * NN_ + p, S + ld_nt(noise + p));
    st_nt(out + 1 * NN_ + p, Vn);
    st_nt(out + 2 * NN_ + p, Rn);
    st_nt(out + 3 * NN_ + p, An);
    st_nt(out + 4 * NN_ + p, Tv);
    st_nt(out + 5 * NN_ + p, Ev);
    st_nt(out + 6 * NN_ + p, eic);
}

extern "C" void kernel_launch(void* const* d_in, const int* in_sizes, int n_in,
                              void* d_out, int out_size, void* d_ws, size_t ws_size,
                              hipStream_t stream) {
    (void)in_sizes; (void)n_in; (void)d_ws; (void)ws_size; (void)out_size;
    const float* x     = (const float*)d_in[0];   // 6*N
    const float* noise = (const float*)d_in[1];   // N
    const float* nn    = (const float*)d_in[2];   // 25*N
    const float* EI    = (const float*)d_in[3];   // N
    const float* kI    = (const float*)d_in[4];   // N
    const float* kE    = (const float*)d_in[5];   // N
    float* out = (float*)d_out;                   // 7*N + 25*N

    dim3 grid(WW / TILE_W, HH / TILE_H);          // 24 x 96
    dim3 block(256);                              // 8 wave32s
    snn_step_kernel<<<grid, block, 0, stream>>>(x, noise, nn, EI, kI, kE, out);
}